// DCT2_1133871366815
// MI455X (gfx1250) — compile-verified
//
#include <hip/hip_runtime.h>

// ---------------------------------------------------------------------------
// 2-D DCT-II (unnormalized) of a 4096x4096 f32 matrix on gfx1250 (MI455X).
//   out = C * X * C^T,  C[k][n] = cos(pi*(2n+1)*k / 8192)
// Two f32 WMMA GEMMs (V_WMMA_F32_16X16X4_F32):
//   pass 1: T = X * C^T   (X staged in LDS, C^T generated in registers)
//   pass 2: out = C * T   (T staged in LDS, C generated in registers)
// CDNA5 specifics:
//   * staged operand uses GLOBAL_LOAD_ASYNC_TO_LDS_B128, double-buffered,
//     synced with s_wait_asynccnt -> global->LDS overlaps the WMMAs
//   * cosine operand generated per-wave straight into WMMA fragment layout
//     with v_cos_f32 (TRANS pipe, co-executes with the XDL/WMMA pipe) after
//     EXACT integer angle reduction: m=(2n+1)k mod 16384, arg m/16384 rev.
// ---------------------------------------------------------------------------

typedef __attribute__((ext_vector_type(2))) float v2f;
typedef __attribute__((ext_vector_type(8))) float v8f;

constexpr int MAT = 4096;
constexpr int BM = 128, BN = 128, BK = 32;
constexpr int AS_STRIDE = BK + 4;    // 36 floats: 16B-aligned rows, bank-skewed
constexpr int BS_STRIDE = BN + 4;    // 132 floats
constexpr int THREADS = 256;         // 8 wave32 waves (4 in M x 2 in N)

// cos(pi*(2n+1)*k/8192). v_cos_f32 takes revolutions: cos(2*pi*x).
// m & 16383 is an exact period reduction; m/16384 is exact in f32.
__device__ __forceinline__ float dct_cos(unsigned n, unsigned k) {
  unsigned m = (((n << 1) + 1u) * k) & 16383u;
  return __builtin_amdgcn_cosf((float)m * (1.0f / 16384.0f));
}

// Async DMA: 16 bytes global -> LDS per lane, tracked by ASYNCcnt.
__device__ __forceinline__ void async_load_b128(unsigned ldsOff, const float* g) {
  asm volatile("global_load_async_to_lds_b128 %0, %1, off"
               :
               : "v"(ldsOff), "v"(g)
               : "memory");
}

// GEN_IS_B = true : pass 1, LDS holds A=X tile (BM x BK), B = cosines.
// GEN_IS_B = false: pass 2, LDS holds B=T tile (BK x BN), A = cosines.
template <bool GEN_IS_B>
__global__ __launch_bounds__(THREADS) void dct_gemm_async(
    const float* __restrict__ G,    // staged operand: X (pass1) or T (pass2)
    float* __restrict__ D) {
  constexpr int STRIDE = GEN_IS_B ? AS_STRIDE : BS_STRIDE;
  constexpr int TILE = GEN_IS_B ? BM * AS_STRIDE : BK * BS_STRIDE;
  __shared__ float Sm[2][TILE];     // double-buffered staged tile

  const int tid = threadIdx.x;
  const int lane = tid & 31;
  const int wave = tid >> 5;
  const int lane16 = lane & 15;
  const int laneHalf = lane >> 4;   // 0: K {0,1}, rows/cols 0-15; 1: K {2,3}, +8

  const int blockM = blockIdx.y * BM;
  const int blockN = blockIdx.x * BN;
  const int wmBase = (wave >> 1) * 32;   // wave tile: 32 (M) x 64 (N)
  const int wnBase = (wave & 1) * 64;

  // per-thread staging pattern: 4 async b128 per thread per tile
  int sRow[4], sC4[4];
#pragma unroll
  for (int i = 0; i < 4; ++i) {
    int idx = tid + i * THREADS;
    if (GEN_IS_B) { sRow[i] = idx >> 3; sC4[i] = (idx & 7) << 2; }   // 128x32
    else          { sRow[i] = idx >> 5; sC4[i] = (idx & 31) << 2; }  // 32x128
  }

  auto stage = [&](int buf, int kk) {
#pragma unroll
    for (int i = 0; i < 4; ++i) {
      const float* g;
      if (GEN_IS_B)
        g = G + (size_t)(blockM + sRow[i]) * MAT + kk + sC4[i];
      else
        g = G + (size_t)(kk + sRow[i]) * MAT + blockN + sC4[i];
      // low 32 bits of the flat shared address == LDS byte offset (ISA 10.2)
      unsigned ldsOff = (unsigned)(uintptr_t)&Sm[buf][sRow[i] * STRIDE + sC4[i]];
      async_load_b128(ldsOff, g);
    }
  };

  v8f acc[2][4] = {};

  stage(0, 0);    // prologue: buffer 0 <- first K tile (4 async ops in flight)

  for (int kk = 0; kk < MAT; kk += BK) {
    const int buf = (kk >> 5) & 1;

    // all waves done reading buf^1 from two steps ago before we overwrite it
    __syncthreads();
    if (kk + BK < MAT) {
      stage(buf ^ 1, kk + BK);                      // prefetch next tile
      asm volatile("s_wait_asynccnt 0x4" ::: "memory");  // current tile landed
    } else {
      asm volatile("s_wait_asynccnt 0x0" ::: "memory");
    }
    __syncthreads();                                // visible to all waves

    // ---- 8 k-steps x 8 wmma = 64 v_wmma_f32_16x16x4_f32 on this tile ----
#pragma unroll 2
    for (int k4 = 0; k4 < BK; k4 += 4) {
      const unsigned kg = (unsigned)(kk + k4 + laneHalf * 2);
      v2f afrag[2], bfrag[4];

      if (GEN_IS_B) {
        // A fragments from LDS (16x4 layout: lanes=M rows, vgpr pair=K)
#pragma unroll
        for (int tm = 0; tm < 2; ++tm)
          afrag[tm] = *(const v2f*)&Sm[buf][(wmBase + tm * 16 + lane16) * STRIDE +
                                            k4 + laneHalf * 2];
        // B fragments generated: B[n][j] = cos(pi*(2n+1)j/8192), n = kg
#pragma unroll
        for (int tn = 0; tn < 4; ++tn) {
          unsigned col = (unsigned)(blockN + wnBase + tn * 16 + lane16);
          v2f b;
          b.x = dct_cos(kg + 0u, col);
          b.y = dct_cos(kg + 1u, col);
          bfrag[tn] = b;
        }
      } else {
        // A fragments generated: A[i][m] = cos(pi*(2m+1)i/8192), m = kg
#pragma unroll
        for (int tm = 0; tm < 2; ++tm) {
          unsigned rowg = (unsigned)(blockM + wmBase + tm * 16 + lane16);
          v2f a;
          a.x = dct_cos(kg + 0u, rowg);
          a.y = dct_cos(kg + 1u, rowg);
          afrag[tm] = a;
        }
        // B fragments from LDS (4x16 layout: lanes=N cols, vgpr pair=K)
#pragma unroll
        for (int tn = 0; tn < 4; ++tn) {
          int col = wnBase + tn * 16 + lane16;
          v2f b;
          b.x = Sm[buf][(k4 + laneHalf * 2 + 0) * STRIDE + col];
          b.y = Sm[buf][(k4 + laneHalf * 2 + 1) * STRIDE + col];
          bfrag[tn] = b;
        }
      }

#pragma unroll
      for (int tm = 0; tm < 2; ++tm)
#pragma unroll
        for (int tn = 0; tn < 4; ++tn)
          acc[tm][tn] = __builtin_amdgcn_wmma_f32_16x16x4_f32(
              false, afrag[tm], false, bfrag[tn], (short)0, acc[tm][tn], false,
              false);
    }
  }

  // ---- store: C/D layout, VGPR r -> row r (lanes 0-15) / r+8 (lanes 16-31)
#pragma unroll
  for (int tm = 0; tm < 2; ++tm) {
#pragma unroll
    for (int tn = 0; tn < 4; ++tn) {
#pragma unroll
      for (int r = 0; r < 8; ++r) {
        int row = blockM + wmBase + tm * 16 + laneHalf * 8 + r;
        int col = blockN + wnBase + tn * 16 + lane16;
        D[(size_t)row * MAT + col] = acc[tm][tn][r];
      }
    }
  }
}

extern "C" void kernel_launch(void* const* d_in, const int* in_sizes, int n_in,
                              void* d_out, int out_size, void* d_ws,
                              size_t ws_size, hipStream_t stream) {
  const float* x = (const float*)d_in[0];
  float* T = (float*)d_ws;            // 4096*4096 f32 = 64 MB intermediate
  float* out = (float*)d_out;

  dim3 grid(MAT / BN, MAT / BM);      // 32 x 32 blocks
  dim3 block(THREADS);

  // Pass 1: T = X * C^T   (B operand generated as cosines)
  dct_gemm_async<true><<<grid, block, 0, stream>>>(x, T);
  // Pass 2: out = C * T   (A operand generated as cosines)
  dct_gemm_async<false><<<grid, block, 0, stream>>>(T, out);
}